// QuantNativeLoss_1159641170364
// MI455X (gfx1250) — compile-verified
//
#include <hip/hip_runtime.h>
#include <cstdint>
#include <cstddef>

#define TLEN      4096
#define NTHREADS  256
#define CHUNK     16          // TLEN / NTHREADS
#define PAD       20          // front zero-pad (80 B -> keeps 16B alignment)
#define HALO      36          // CHUNK + PAD halo span held in registers
#define NACC      11
#define NWAVES    (NTHREADS / 32)

#define W_IC   0.4f
#define W_VOL  0.3f
#define W_NORM 0.2f
#define W_MOM  0.07f
#define W_DD   0.03f
#define ALPHA  0.5f
#define EPSF   1e-8f

__device__ __forceinline__ float wave_sum(float v) {
#pragma unroll
  for (int off = 16; off > 0; off >>= 1) v += __shfl_down(v, off, 32);
  return v;
}

__global__ __launch_bounds__(NTHREADS)
void quant_loss_row_kernel(const float* __restrict__ pred,
                           const float* __restrict__ target,
                           float* __restrict__ partial,
                           float invB, float invBT) {
  __shared__ __align__(16) float sp[PAD + TLEN];
  __shared__ __align__(16) float st[PAD + TLEN];
  __shared__ int   wtot[NWAVES];
  __shared__ float red[NWAVES][NACC];

  const int row  = blockIdx.x;
  const int tid  = threadIdx.x;
  const int lane = tid & 31;
  const int wave = tid >> 5;
  const float* __restrict__ prow = pred   + (size_t)row * TLEN;
  const float* __restrict__ trow = target + (size_t)row * TLEN;

  // ---- Stage full row into LDS via CDNA5 async global->LDS DMA path ------
#if defined(__gfx1250__)
  {
#pragma unroll
    for (int i = 0; i < 4; ++i) {
      unsigned idx  = (unsigned)(tid * 4 + i * (NTHREADS * 4)); // float index
      unsigned goff = idx * 4u;                                 // byte offset
      unsigned lp = (unsigned)(uintptr_t)(&sp[PAD + idx]);      // 16B aligned
      unsigned lt = (unsigned)(uintptr_t)(&st[PAD + idx]);
      asm volatile("global_load_async_to_lds_b128 %0, %1, %2"
                   :: "v"(lp), "v"(goff), "s"(prow) : "memory");
      asm volatile("global_load_async_to_lds_b128 %0, %1, %2"
                   :: "v"(lt), "v"(goff), "s"(trow) : "memory");
    }
    if (tid < PAD) { sp[tid] = 0.0f; st[tid] = 0.0f; }          // zero halo pad
    asm volatile("s_wait_asynccnt 0x0" ::: "memory");
  }
#else
  for (int i = tid; i < TLEN; i += NTHREADS) {
    sp[PAD + i] = prow[i]; st[PAD + i] = trow[i];
  }
  if (tid < PAD) { sp[tid] = 0.0f; st[tid] = 0.0f; }
#endif
  __syncthreads();

  const int base = tid * CHUNK;  // LDS index `base + k` holds element base-PAD+k

  // ---- Pull the 36-float halo [base-20, base+15] into registers ----------
  // element t lives at h[(t - base) + 20]; all loads are aligned b128.
  float hp[HALO], hx[HALO];
#pragma unroll
  for (int k = 0; k < HALO; k += 4) {
    const float4 vp = *(const float4*)(&sp[base + k]);
    const float4 vx = *(const float4*)(&st[base + k]);
    hp[k] = vp.x; hp[k + 1] = vp.y; hp[k + 2] = vp.z; hp[k + 3] = vp.w;
    hx[k] = vx.x; hx[k + 1] = vx.y; hx[k + 2] = vx.z; hx[k + 3] = vx.w;
  }

  // ---- Window warm-up from the (zero-padded) preceding elements ----------
  float w20p = 0.f, w20t = 0.f, s1t10 = 0.f, s2t10 = 0.f;
#pragma unroll
  for (int k = 1; k < 20; ++k) {                 // elements base-19 .. base-1
    w20p += hp[k] * hp[k];  w20t += hx[k] * hx[k];
  }
#pragma unroll
  for (int k = 11; k < 20; ++k) {                // elements base-9 .. base-1
    s1t10 += hx[k];  s2t10 += hx[k] * hx[k];
  }

  // ---- Pass A: moments + windowed terms + local "last nonneg" ------------
  float a_sp = 0.f, a_sp2 = 0.f, a_sp3 = 0.f;
  float a_st = 0.f, a_st2 = 0.f, a_st3 = 0.f, a_spt = 0.f;
  float a_vol = 0.f, a_norm = 0.f;
  int   mloc = -1;

  // Per-element body; rn20 = 1/n20, rn10 = 1/n10, rd10 = 1/max(n10-1,1).
  auto elem = [&](int i, float rn20, float rn10, float rd10) {
    const int t = base + i;
    const float p = hp[20 + i];
    const float x = hx[20 + i];
    const float d = p - x;

    a_sp  += p;  a_sp2 += p * p;  a_sp3 += p * p * p;
    a_st  += x;  a_st2 += x * x;  a_st3 += x * x * x;
    a_spt += p * x;

    // realized vol, trailing window 20; single log of the RV ratio
    w20p += p * p;  w20t += x * x;
    const float rvp = sqrtf(w20p * rn20) + EPSF;
    const float rvt = sqrtf(w20t * rn20) + EPSF;
    const float lv  = logf(rvp / rvt);
    a_vol += lv * lv;
    w20p -= hp[i + 1] * hp[i + 1];   // element t-19 (zero inside the pad)
    w20t -= hx[i + 1] * hx[i + 1];

    // rolling unbiased std of target, window 10
    s1t10 += x;  s2t10 += x * x;
    const float var = (s2t10 - s1t10 * s1t10 * rn10) * rd10;
    const float sig = sqrtf(fmaxf(var, 0.0f)) + EPSF;
    const float r   = d / sig;
    a_norm += r * r;
    s1t10 -= hx[i + 11];             // element t-9 (zero inside the pad)
    s2t10 -= hx[i + 11] * hx[i + 11];

    if (x >= 0.0f) mloc = t;         // last non-negative index in this chunk
  };

  if (base >= 19) {
    // Steady state (254 of 256 threads; 7 of 8 waves take only this path):
    // window counts are the constants 20 / 10 / 9.
#pragma unroll
    for (int i = 0; i < CHUNK; ++i)
      elem(i, 1.0f / 20.0f, 1.0f / 10.0f, 1.0f / 9.0f);
  } else {
    // Row warm-up (threads 0..1 only): expanding windows, exact divides.
#pragma unroll
    for (int i = 0; i < CHUNK; ++i) {
      const int t = base + i;
      const float n20 = fminf((float)(t + 1), 20.0f);
      const float n10 = fminf((float)(t + 1), 10.0f);
      elem(i, 1.0f / n20, 1.0f / n10, 1.0f / fmaxf(n10 - 1.0f, 1.0f));
    }
  }

  // ---- Exclusive max-scan of "last nonneg index" (wave32 shfl + 1 barrier)
  int incl = mloc;
#pragma unroll
  for (int off = 1; off < 32; off <<= 1) {
    const int u = __shfl_up(incl, off, 32);
    if (lane >= off && u > incl) incl = u;
  }
  if (lane == 31) wtot[wave] = incl;
  __syncthreads();
  int wpref = -1;
#pragma unroll
  for (int w = 0; w < NWAVES; ++w) {
    const int v = wtot[w];
    if (w < wave && v > wpref) wpref = v;
  }
  const int prev = __shfl_up(incl, 1, 32);       // inclusive of lane-1
  int lastpref = wpref;
  if (lane > 0 && prev > lastpref) lastpref = prev;

  // ---- Pass B: drawdown-weighted residual (all from registers) -----------
  float a_wd2 = 0.f, a_w = 0.f;
  int last = lastpref;
#pragma unroll
  for (int i = 0; i < CHUNK; ++i) {
    const int t = base + i;
    const float x = hx[20 + i];
    float streak;
    if (x >= 0.0f) { last = t; streak = 0.0f; }
    else           { streak = (float)(t - last); }
    const float w = sqrtf(streak + 1.0f);
    const float d = hp[20 + i] - x;
    a_wd2 += w * d * d;
    a_w   += w;
  }

  // ---- Deterministic block reduction of 11 accumulators ------------------
  float acc[NACC] = { a_sp, a_sp2, a_sp3, a_st, a_st2, a_st3, a_spt,
                      a_vol, a_norm, a_wd2, a_w };
#pragma unroll
  for (int k = 0; k < NACC; ++k) {
    const float v = wave_sum(acc[k]);
    if (lane == 0) red[wave][k] = v;
  }
  __syncthreads();

  if (tid == 0) {
    float s[NACC];
#pragma unroll
    for (int k = 0; k < NACC; ++k) {
      float v = 0.f;
#pragma unroll
      for (int w = 0; w < NWAVES; ++w) v += red[w][k];
      s[k] = v;
    }
    const float Tn = (float)TLEN;
    const float mp = s[0] / Tn, mt = s[3] / Tn;
    const float cp2 = fmaxf(s[1] - Tn * mp * mp, 0.0f);   // sum centered sq
    const float ct2 = fmaxf(s[4] - Tn * mt * mt, 0.0f);
    const float num = s[6] - Tn * mp * mt;
    const float corr = num / (sqrtf(cp2) * sqrtf(ct2) + EPSF);
    const float l_ic = 1.0f - corr;

    const float stdp = sqrtf(cp2 / (Tn - 1.0f)) + EPSF;   // ddof=1
    const float stdt = sqrtf(ct2 / (Tn - 1.0f)) + EPSF;
    const float rel  = (stdp - stdt) / stdt;
    const float l_var = rel * rel;

    // sum (x-mu)^3 = Sx3 - 3 mu Sx2 + 2 T mu^3
    const float c3p = s[2] - 3.0f * mp * s[1] + 2.0f * Tn * mp * mp * mp;
    const float c3t = s[5] - 3.0f * mt * s[4] + 2.0f * Tn * mt * mt * mt;
    const float skp = c3p / (Tn * stdp * stdp * stdp);
    const float skt = c3t / (Tn * stdt * stdt * stdt);
    const float dsk = skp - skt;
    const float l_skew = dsk * dsk;

    // sum_t [w/(mean_w+eps)] d^2  ==  (sum w d^2) / (mean_w + eps)
    const float dd = s[9] / (s[10] / Tn + EPSF);

    partial[row] = W_IC   * l_ic  * invB
                 + W_VOL  * s[7]  * invBT
                 + W_NORM * s[8]  * invBT
                 + W_MOM  * ((1.0f - ALPHA) * l_var + ALPHA * l_skew) * invB
                 + W_DD   * dd    * invBT;
  }
}

__global__ __launch_bounds__(256)
void reduce_partials_kernel(const float* __restrict__ partial, int n,
                            float* __restrict__ out) {
  __shared__ float s[256];
  float v = 0.f;
  for (int i = threadIdx.x; i < n; i += 256) v += partial[i]; // fixed order
  s[threadIdx.x] = v;
  __syncthreads();
  if (threadIdx.x == 0) {
    float tot = 0.f;
#pragma unroll 8
    for (int i = 0; i < 256; ++i) tot += s[i];               // fixed order
    out[0] = tot;
  }
}

extern "C" void kernel_launch(void* const* d_in, const int* in_sizes, int n_in,
                              void* d_out, int out_size, void* d_ws, size_t ws_size,
                              hipStream_t stream) {
  const float* pred   = (const float*)d_in[0];
  const float* target = (const float*)d_in[1];
  const int B = in_sizes[0] / TLEN;

  float* partial = (float*)d_ws;
  const float invB  = 1.0f / (float)B;
  const float invBT = 1.0f / ((float)B * (float)TLEN);

  quant_loss_row_kernel<<<B, NTHREADS, 0, stream>>>(pred, target, partial,
                                                    invB, invBT);
  reduce_partials_kernel<<<1, 256, 0, stream>>>(partial, B, (float*)d_out);
}